// SemanticConditioner_54778012893648
// MI455X (gfx1250) — compile-verified
//
#include <hip/hip_runtime.h>

// ---------------------------------------------------------------------------
// SemanticConditioner for MI455X (gfx1250, wave32)
//
//   all_cond[r,d] = sum_e embeddings[r,e] * W[d,e] + residuals[r,d]   (R=512, D=256, E=1536)
//   out[b,n,d]    = canvas[b,n,d] + all_cond[region_ids[n], d]        (B=4, N=65536)
//
// Stage 1 is a tiny fp32 GEMM (~0.4 GFLOP) -> V_WMMA_F32_16X16X4_F32 (exact fp32).
// Stage 2 is the real cost: ~537 MB of HBM traffic => ~23 us at 23.3 TB/s.
// all_cond (512 KB) + region_ids (256 KB) stay resident in the 192 MB L2.
// ---------------------------------------------------------------------------

typedef __attribute__((ext_vector_type(2))) float v2f;
typedef __attribute__((ext_vector_type(8))) float v8f;

constexpr int Bc = 4;
constexpr int Nc = 65536;
constexpr int Dc = 256;
constexpr int Ec = 1536;
constexpr int Rc = 512;

// ---------------------------------------------------------------------------
// Stage 1: all_cond = embeddings @ W^T + residuals, via V_WMMA_F32_16X16X4_F32.
// One 16x16 output tile per block; 4 waves split K=1536 into 384-element
// chunks (96 WMMAs each), then reduce partial accumulators through LDS.
//
// Fragment layouts (ISA 7.12.2, 32-bit, wave32):
//   A 16x4 (MxK):  lane L holds row M=L%16; VGPR0 -> K = 0 (lanes 0-15) / 2 (16-31),
//                  VGPR1 -> K = 1 / 3.
//   B 4x16 (KxN):  lane L holds col N=L%16; VGPR0 -> K = 0 / 2, VGPR1 -> K = 1 / 3.
//   C/D 16x16:     VGPR i -> row M = i (lanes 0-15) / i+8 (lanes 16-31), col N=L%16.
// ---------------------------------------------------------------------------
__global__ __launch_bounds__(128)
void cond_gemm_kernel(const float* __restrict__ emb,     // [R, E]
                      const float* __restrict__ W,       // [D, E]
                      const float* __restrict__ resid,   // [R, D]
                      float* __restrict__ all_cond)      // [R, D]
{
    __shared__ float red[4][32 * 8];

    const int tid  = threadIdx.x;
    const int lane = tid & 31;
    const int wave = tid >> 5;
    const int half = lane >> 4;   // 0: lanes 0-15, 1: lanes 16-31
    const int l    = lane & 15;

    const int tile   = blockIdx.x;            // 0 .. (R/16)*(D/16)-1 = 511
    const int tile_r = (tile >> 4) << 4;      // R tile base (32 tiles of 16)
    const int tile_d = (tile & 15) << 4;      // D tile base (16 tiles of 16)

    v8f c = {0.f, 0.f, 0.f, 0.f, 0.f, 0.f, 0.f, 0.f};

    // This wave covers K in [kbase, kbase + E/4)
    const int kbase = wave * (Ec / 4);
    const float* __restrict__ arow = emb + (size_t)(tile_r + l) * Ec;  // A row M=l
    const float* __restrict__ brow = W   + (size_t)(tile_d + l) * Ec;  // B col N=l (W row)

    #pragma unroll 4
    for (int kk = 0; kk < Ec / 4; kk += 4) {
        const int ka = kbase + kk + 2 * half;
        v2f a, b;
        a.x = arow[ka];
        a.y = arow[ka + 1];
        b.x = brow[ka];
        b.y = brow[ka + 1];
        // 8 args: (neg_a, A, neg_b, B, c_mod, C, reuse_a, reuse_b)
        c = __builtin_amdgcn_wmma_f32_16x16x4_f32(false, a, false, b,
                                                  (short)0, c, false, false);
    }

    // Reduce the 4 partial accumulators through LDS.
    #pragma unroll
    for (int i = 0; i < 8; ++i) red[wave][lane * 8 + i] = c[i];
    __syncthreads();

    if (wave == 0) {
        #pragma unroll
        for (int i = 0; i < 8; ++i) {
            const float s = red[0][lane * 8 + i] + red[1][lane * 8 + i] +
                            red[2][lane * 8 + i] + red[3][lane * 8 + i];
            const int row = tile_r + i + 8 * half;   // C/D layout: VGPR i -> M=i / i+8
            const int col = tile_d + l;
            all_cond[(size_t)row * Dc + col] = s + resid[(size_t)row * Dc + col];
        }
    }
}

// ---------------------------------------------------------------------------
// Stage 2: out[b,n,:] = canvas[b,n,:] + all_cond[region_ids[n],:]
// 256 threads/block = 4 rows of D=256 floats as float4 (64 float4 per row).
// Each thread gathers its cond float4 ONCE (L2-resident), then streams all
// B=4 batches: fully coalesced b128 loads/stores, 4x reuse of the gather.
// ---------------------------------------------------------------------------
__global__ __launch_bounds__(256)
void broadcast_add_kernel(const float4* __restrict__ canvas,   // [B*N*D/4]
                          const float4* __restrict__ all_cond, // [R*D/4]
                          const int* __restrict__ region_ids,  // [N]
                          float4* __restrict__ out)            // [B*N*D/4]
{
    const int tid = threadIdx.x;
    const int row = blockIdx.x * 4 + (tid >> 6);   // n position, 4 rows per block
    const int q   = tid & 63;                      // float4 index within D

    const int rid = region_ids[row];
    const float4 cond = all_cond[(size_t)rid * (Dc / 4) + q];

    const size_t base = (size_t)row * (Dc / 4) + q;
    constexpr size_t batch_stride = (size_t)Nc * (Dc / 4);

    #pragma unroll
    for (int b = 0; b < Bc; ++b) {
        const size_t idx = (size_t)b * batch_stride + base;
        const float4 cv = canvas[idx];
        float4 o;
        o.x = cv.x + cond.x;
        o.y = cv.y + cond.y;
        o.z = cv.z + cond.z;
        o.w = cv.w + cond.w;
        out[idx] = o;
    }
}

// ---------------------------------------------------------------------------
// Launch
// ---------------------------------------------------------------------------
extern "C" void kernel_launch(void* const* d_in, const int* in_sizes, int n_in,
                              void* d_out, int out_size, void* d_ws, size_t ws_size,
                              hipStream_t stream) {
    const float* canvas     = (const float*)d_in[0];  // [B, N, D]
    const float* embeddings = (const float*)d_in[1];  // [R, E]
    const float* W          = (const float*)d_in[2];  // [D, E]
    const float* residuals  = (const float*)d_in[3];  // [R, D]
    const int*   region_ids = (const int*)d_in[4];    // [N]
    float*       out        = (float*)d_out;          // [B, N, D]

    float* all_cond = (float*)d_ws;                   // [R, D] = 512 KB scratch

    // Stage 1: 512 tiles (32 x 16), 4 waves each.
    cond_gemm_kernel<<<dim3((Rc / 16) * (Dc / 16)), dim3(128), 0, stream>>>(
        embeddings, W, residuals, all_cond);

    // Stage 2: N/4 blocks x 256 threads, float4 streaming.
    broadcast_add_kernel<<<dim3(Nc / 4), dim3(256), 0, stream>>>(
        (const float4*)canvas, (const float4*)all_cond, region_ids,
        (float4*)out);
}